// ImagePiecewiseRigid_56324201120126
// MI455X (gfx1250) — compile-verified
//
#include <hip/hip_runtime.h>
#include <math.h>

// ---------------------------------------------------------------------------
// ImagePiecewiseRigid forward for MI455X (gfx1250, wave32).
// Everything fp32 (reference dtype). All GEMM-shaped work (conv1, conv2, FC1)
// goes through V_WMMA_F32_16X16X4_F32. Intermediates live in d_ws (~10 MB)
// so the whole pipeline is L2-resident (192 MB L2, 23.3 TB/s HBM never the
// bottleneck; this is launch/latency bound -> aggressive fusion).
// ---------------------------------------------------------------------------

typedef float v2f __attribute__((ext_vector_type(2)));
typedef float v8f __attribute__((ext_vector_type(8)));

// D = A(16x4) * B(4x16) + C, fp32.  Assumed lane layout (ISA 7.12.2):
//   A: lane = half*16+m, vgpr v holds A[m][2*half+v]
//   B: lane = half*16+n, vgpr v holds B[2*half+v][n]
//   C/D: lane = half*16+n, vgpr v holds C[v+8*half][n]
__device__ __forceinline__ v8f wmma4(v2f a, v2f b, v8f c) {
  return __builtin_amdgcn_wmma_f32_16x16x4_f32(false, a, false, b, (short)0, c,
                                               false, false);
}

__constant__ float c_tess[8][3] = {
    {32.f, 32.f, 4.f}, {32.f, 96.f, 4.f}, {96.f, 32.f, 4.f}, {96.f, 96.f, 4.f},
    {32.f, 64.f, 4.f}, {96.f, 64.f, 4.f}, {64.f, 32.f, 4.f}, {64.f, 96.f, 4.f}};

// ---------------------------------------------------------------------------
// Kernel 1: conv1(4->8ch, 7x7x1) + bias + 2x2 maxpool(D,H) + relu, fused.
// Implicit GEMM: M-tile = 16 output positions = (h-pair)x(8 w) at fixed d.
// One wave computes the two d-tiles of a pool cell -> pooled in registers.
// x: (8,4,128,128,8)  -> h1: (8,8,61,61,8)
// ---------------------------------------------------------------------------
__global__ void k_conv1_pool(const float* __restrict__ x,
                             const float* __restrict__ w1,
                             const float* __restrict__ b1,
                             float* __restrict__ h1) {
  __shared__ float Bs[196 * 16];
  __shared__ int offs[196];
  const int dp = blockIdx.x;  // 0..60
  const int b = blockIdx.y;   // 0..7
  const int tid = threadIdx.x;
  for (int idx = tid; idx < 196 * 16; idx += 256) {
    int k = idx >> 4, n = idx & 15;
    int ic = k / 49, r = k % 49, kd = r / 7, kh = r % 7;
    Bs[idx] = (n < 8) ? w1[((n * 4 + ic) * 7 + kd) * 7 + kh] : 0.0f;
  }
  for (int k = tid; k < 196; k += 256) {
    int ic = k / 49, r = k % 49, kd = r / 7, kh = r % 7;
    offs[k] = ic * (128 * 128 * 8) + kd * (128 * 8) + kh * 8;
  }
  __syncthreads();
  const int wave = tid >> 5, lane = tid & 31;
  const int hf = lane >> 4, mm = lane & 15;
  const int hoff = mm >> 3, ww = mm & 7;
  const float bias = b1[mm & 7];
  for (int hp = wave; hp < 61; hp += 8) {
    const int h = 2 * hp + hoff;
    int base0 = (((b * 4) * 128 + (2 * dp + 0)) * 128 + h) * 8 + ww;
    int base1 = base0 + 128 * 8;  // d+1
    v8f c0 = {}, c1 = {};
    for (int s = 0; s < 49; ++s) {
      int k0 = 4 * s + 2 * hf;
      int o0 = offs[k0], o1 = offs[k0 + 1];
      v2f a0, a1, bb;
      a0.x = x[base0 + o0]; a0.y = x[base0 + o1];
      a1.x = x[base1 + o0]; a1.y = x[base1 + o1];
      bb.x = Bs[k0 * 16 + mm]; bb.y = Bs[(k0 + 1) * 16 + mm];
      c0 = wmma4(a0, bb, c0);
      c1 = wmma4(a1, bb, c1);
    }
    float out[8];
#pragma unroll
    for (int v = 0; v < 8; ++v) {
      float p0 = fmaxf(c0[v], __shfl_xor(c0[v], 16, 32));  // pool over h-pair
      float p1 = fmaxf(c1[v], __shfl_xor(c1[v], 16, 32));
      out[v] = fmaxf(fmaxf(p0, p1) + bias, 0.0f);  // pool over d-pair + relu
    }
    if (lane < 8) {  // lane = out channel, v = w
      float* dst = &h1[(((b * 8 + lane) * 61 + dp) * 61 + hp) * 8];
#pragma unroll
      for (int v = 0; v < 8; ++v) dst[v] = out[v];
    }
  }
}

// ---------------------------------------------------------------------------
// Kernel 2: conv2(8->10ch, 5x5x1) + bias + pool + relu. Same scheme, K=200.
// h1: (8,8,61,61,8) -> feat: (8, 62720) in reshape order ((c*28+d)*28+h)*8+w
// ---------------------------------------------------------------------------
__global__ void k_conv2_pool(const float* __restrict__ h1,
                             const float* __restrict__ w2,
                             const float* __restrict__ b2,
                             float* __restrict__ feat) {
  __shared__ float Bs[200 * 16];
  __shared__ int offs[200];
  const int dp = blockIdx.x;  // 0..27
  const int b = blockIdx.y;
  const int tid = threadIdx.x;
  for (int idx = tid; idx < 200 * 16; idx += 256) {
    int k = idx >> 4, n = idx & 15;
    int ic = k / 25, r = k % 25, kd = r / 5, kh = r % 5;
    Bs[idx] = (n < 10) ? w2[((n * 8 + ic) * 5 + kd) * 5 + kh] : 0.0f;
  }
  for (int k = tid; k < 200; k += 256) {
    int ic = k / 25, r = k % 25, kd = r / 5, kh = r % 5;
    offs[k] = ic * (61 * 61 * 8) + kd * (61 * 8) + kh * 8;
  }
  __syncthreads();
  const int wave = tid >> 5, lane = tid & 31;
  const int hf = lane >> 4, mm = lane & 15;
  const int hoff = mm >> 3, ww = mm & 7;
  const float bias = b2[mm < 10 ? mm : 0];
  for (int hp = wave; hp < 28; hp += 8) {
    const int h = 2 * hp + hoff;
    int base0 = (((b * 8) * 61 + (2 * dp + 0)) * 61 + h) * 8 + ww;
    int base1 = base0 + 61 * 8;
    v8f c0 = {}, c1 = {};
    for (int s = 0; s < 50; ++s) {
      int k0 = 4 * s + 2 * hf;
      int o0 = offs[k0], o1 = offs[k0 + 1];
      v2f a0, a1, bb;
      a0.x = h1[base0 + o0]; a0.y = h1[base0 + o1];
      a1.x = h1[base1 + o0]; a1.y = h1[base1 + o1];
      bb.x = Bs[k0 * 16 + mm]; bb.y = Bs[(k0 + 1) * 16 + mm];
      c0 = wmma4(a0, bb, c0);
      c1 = wmma4(a1, bb, c1);
    }
    float out[8];
#pragma unroll
    for (int v = 0; v < 8; ++v) {
      float p0 = fmaxf(c0[v], __shfl_xor(c0[v], 16, 32));
      float p1 = fmaxf(c1[v], __shfl_xor(c1[v], 16, 32));
      out[v] = fmaxf(fmaxf(p0, p1) + bias, 0.0f);
    }
    if (lane < 10) {
      float* dst = &feat[b * 62720 + ((lane * 28 + dp) * 28 + hp) * 8];
#pragma unroll
      for (int v = 0; v < 8; ++v) dst[v] = out[v];
    }
  }
}

// ---------------------------------------------------------------------------
// FC1: (8 x 62720) @ (62720 x 32). Split-K WMMA: 2 N-tiles x 16 K-chunks.
// ---------------------------------------------------------------------------
__global__ void k_fc1_init(const float* __restrict__ fc1b, float* fc1out) {
  int i = threadIdx.x;  // 256
  fc1out[i] = fc1b[i & 31];
}

__global__ void k_fc1_wmma(const float* __restrict__ feat,
                           const float* __restrict__ fc1w,
                           float* __restrict__ fc1out) {
  const int ntile = blockIdx.x >> 4;   // 0..1
  const int kchunk = blockIdx.x & 15;  // 0..15
  const int lane = threadIdx.x;
  const int hf = lane >> 4, mm = lane & 15;
  const int n = ntile * 16 + mm;
  const float mvalid = (mm < 8) ? 1.0f : 0.0f;  // mask, keeps EXEC full
  const float* arow = feat + (mm < 8 ? mm : 0) * 62720;
  const float* brow = fc1w + n * 62720;
  const int kbase = kchunk * 3920 + 2 * hf;
  v8f c = {};
  for (int s = 0; s < 980; ++s) {
    int k = kbase + 4 * s;
    v2f a, bb;
    a.x = arow[k] * mvalid;
    a.y = arow[k + 1] * mvalid;
    bb.x = brow[k];
    bb.y = brow[k + 1];
    c = wmma4(a, bb, c);
  }
  if (hf == 0) {  // vgpr v holds row m=v (0..7 = the 8 batches)
#pragma unroll
    for (int v = 0; v < 8; ++v) atomicAdd(&fc1out[v * 32 + n], c[v]);
  }
}

// ---------------------------------------------------------------------------
// FC2 + tanh + rigid transform. One thread per (b,t). RT: [b][t][R(9),T(3)]
// ---------------------------------------------------------------------------
__global__ void k_fc2_rigid(const float* __restrict__ fc1out,
                            const float* __restrict__ fc2w,
                            const float* __restrict__ fc2b,
                            float* __restrict__ RT) {
  const int tid = threadIdx.x;  // 64
  const int b = tid >> 3, t = tid & 7;
  float th[6];
#pragma unroll
  for (int j = 0; j < 6; ++j) {
    int o = t * 6 + j;
    float acc = fc2b[o];
    for (int i = 0; i < 32; ++i)
      acc += fmaxf(fc1out[b * 32 + i], 0.0f) * fc2w[o * 32 + i];
    th[j] = tanhf(acc);
  }
  const float PI = 3.14159265358979323846f;
  float a0 = PI * th[0], a1 = PI * th[1], a2 = PI * th[2];
  float c0 = cosf(a0), s0 = sinf(a0);
  float c1 = cosf(a1), s1 = sinf(a1);
  float c2 = cosf(a2), s2 = sinf(a2);
  float R[9];  // R = Z(a0) @ Y(a1) @ X(a2), row-major
  R[0] = c0 * c1;  R[1] = -s0 * c2 + c0 * s1 * s2;  R[2] = s0 * s2 + c0 * s1 * c2;
  R[3] = s0 * c1;  R[4] =  c0 * c2 + s0 * s1 * s2;  R[5] = -c0 * s2 + s0 * s1 * c2;
  R[6] = -s1;      R[7] =  c1 * s2;                 R[8] = c1 * c2;
  const float cen[3] = {64.f, 64.f, 4.f};
  const float szv[3] = {128.f, 128.f, 8.f};
  float* dst = &RT[(b * 8 + t) * 12];
#pragma unroll
  for (int i = 0; i < 9; ++i) dst[i] = R[i];
#pragma unroll
  for (int k = 0; k < 3; ++k)
    dst[9 + k] = th[3 + k] * szv[k] + cen[k] -
                 (cen[0] * R[k] + cen[1] * R[3 + k] + cen[2] * R[6 + k]);
}

// ---------------------------------------------------------------------------
// Flow field -> normalized grid. grid[...,:] = [0, nf1, nf0].
// ---------------------------------------------------------------------------
__global__ void k_grid(const float* __restrict__ RT, float* __restrict__ grd) {
  __shared__ float sRT[96];
  const int b = blockIdx.y;
  const int tid = threadIdx.x;
  if (tid < 96) sRT[tid] = RT[b * 96 + tid];
  __syncthreads();
  const int l = blockIdx.x * 256 + tid;  // 0..131071
  const float gz = (float)(l & 7);
  const float gy = (float)((l >> 3) & 127);
  const float gx = (float)(l >> 10);
  float wt[8], wsum = 0.f;
#pragma unroll
  for (int t = 0; t < 8; ++t) {
    float dx = gx - c_tess[t][0], dy = gy - c_tess[t][1], dz = gz - c_tess[t][2];
    wt[t] = expf(-sqrtf(dx * dx + dy * dy + dz * dz) * 0.1f);
    wsum += wt[t];
  }
  const float inv = 1.0f / wsum;
  float acc0 = 0.f, acc1 = 0.f;
#pragma unroll
  for (int t = 0; t < 8; ++t) {
    const float* r = &sRT[t * 12];
    float f0 = gx * r[0] + gy * r[3] + gz * r[6] + r[9];
    float f1 = gx * r[1] + gy * r[4] + gz * r[7] + r[10];
    float w = wt[t] * inv;
    acc0 += w * f0;
    acc1 += w * f1;
  }
  float nf0 = 2.0f * acc0 * (1.0f / 128.0f) - 1.0f;
  float nf1 = 2.0f * acc1 * (1.0f / 128.0f) - 1.0f;
  float* g = grd + (size_t)(b * 131072 + l) * 3;
  g[0] = 0.0f;
  g[1] = nf1;
  g[2] = nf0;
}

// ---------------------------------------------------------------------------
// Trilinear grid-sample of x(8,4,128,128,8) at grid -> X_t.
// ---------------------------------------------------------------------------
__global__ void k_sample(const float* __restrict__ x,
                         const float* __restrict__ grd,
                         float* __restrict__ Xt) {
  const int b = blockIdx.y;
  const int l = blockIdx.x * 256 + threadIdx.x;
  const float* g = grd + (size_t)(b * 131072 + l) * 3;
  const float ix = ((g[0] + 1.f) * 8.f - 1.f) * 0.5f;
  const float iy = ((g[1] + 1.f) * 128.f - 1.f) * 0.5f;
  const float iz = ((g[2] + 1.f) * 128.f - 1.f) * 0.5f;
  const float z0 = floorf(iz), y0 = floorf(iy), x0 = floorf(ix);
  float acc[4] = {0.f, 0.f, 0.f, 0.f};
#pragma unroll
  for (int dz = 0; dz < 2; ++dz)
#pragma unroll
    for (int dy = 0; dy < 2; ++dy)
#pragma unroll
      for (int dx = 0; dx < 2; ++dx) {
        float zc = z0 + dz, yc = y0 + dy, xc = x0 + dx;
        float w = (1.f - fabsf(iz - zc)) * (1.f - fabsf(iy - yc)) *
                  (1.f - fabsf(ix - xc));
        bool valid = (zc >= 0.f) && (zc < 128.f) && (yc >= 0.f) &&
                     (yc < 128.f) && (xc >= 0.f) && (xc < 8.f);
        float wv = valid ? w : 0.f;
        int zi = (int)fminf(fmaxf(zc, 0.f), 127.f);
        int yi = (int)fminf(fmaxf(yc, 0.f), 127.f);
        int xi = (int)fminf(fmaxf(xc, 0.f), 7.f);
        size_t base = ((size_t)((b * 4) * 128 + zi) * 128 + yi) * 8 + xi;
#pragma unroll
        for (int c = 0; c < 4; ++c) acc[c] += wv * x[base + (size_t)c * 131072];
      }
#pragma unroll
  for (int c = 0; c < 4; ++c)
    Xt[(size_t)(b * 4 + c) * 131072 + l] = acc[c];
}

// ---------------------------------------------------------------------------
// moved + reg (LDS reduction). rintf == round-half-even == jnp.round.
// ---------------------------------------------------------------------------
__global__ void k_moved_reg(const float* __restrict__ pos,
                            const float* __restrict__ centers,
                            const float* __restrict__ grd,
                            float* __restrict__ movedOut,
                            float* __restrict__ regOut) {
  __shared__ float red[128];
  const int b = blockIdx.x, p = threadIdx.x;
  float nrm = 0.f;
  if (p < 100) {
    const float* pp = pos + (b * 100 + p) * 3;
    const float szm1[3] = {127.f, 127.f, 7.f};
    const float szv[3] = {128.f, 128.f, 8.f};
    int idx[3];
#pragma unroll
    for (int c = 0; c < 3; ++c)
      idx[c] = (int)fminf(fmaxf(rintf(pp[c]), 0.f), szm1[c]);
    const float* g =
        grd + ((size_t)b * 131072 + ((idx[0] * 128 + idx[1]) * 8 + idx[2])) * 3;
    const float a[3] = {g[2], g[1], 0.0f};  // gridback = grid[..., ::-1]
    float s = 0.f;
#pragma unroll
    for (int c = 0; c < 3; ++c) {
      float m = 2.f * pp[c] - (0.5f + 0.5f * a[c]) * (szv[c] - 1.f);
      movedOut[(b * 100 + p) * 3 + c] = m;
      float d = m - centers[p * 3 + c];
      s += d * d;
    }
    nrm = sqrtf(s);
  }
  red[p] = nrm;
  __syncthreads();
  for (int off = 64; off > 0; off >>= 1) {
    if (p < off) red[p] += red[p + off];
    __syncthreads();
  }
  if (p == 0) regOut[b] = red[0] * 0.01f;
}

// ---------------------------------------------------------------------------
extern "C" void kernel_launch(void* const* d_in, const int* in_sizes, int n_in,
                              void* d_out, int out_size, void* d_ws,
                              size_t ws_size, hipStream_t stream) {
  (void)in_sizes; (void)n_in; (void)out_size; (void)ws_size;
  const float* x = (const float*)d_in[0];
  const float* pos = (const float*)d_in[1];
  const float* centers = (const float*)d_in[2];
  const float* conv1_w = (const float*)d_in[3];
  const float* conv1_b = (const float*)d_in[4];
  const float* conv2_w = (const float*)d_in[5];
  const float* conv2_b = (const float*)d_in[6];
  const float* fc1_w = (const float*)d_in[7];
  const float* fc1_b = (const float*)d_in[8];
  const float* fc2_w = (const float*)d_in[9];
  const float* fc2_b = (const float*)d_in[10];

  float* out = (float*)d_out;  // X_t | grid | reg | moved
  float* Xt = out;
  float* grd = out + 4194304;
  float* reg = out + 7340032;
  float* moved = out + 7340040;

  float* ws = (float*)d_ws;
  float* h1 = ws;                  // 1,905,152
  float* feat = ws + 1905152;      //   501,760
  float* fc1out = ws + 2406912;    //       256
  float* RT = ws + 2407168;        //       768

  k_conv1_pool<<<dim3(61, 8), 256, 0, stream>>>(x, conv1_w, conv1_b, h1);
  k_conv2_pool<<<dim3(28, 8), 256, 0, stream>>>(h1, conv2_w, conv2_b, feat);
  k_fc1_init<<<1, 256, 0, stream>>>(fc1_b, fc1out);
  k_fc1_wmma<<<32, 32, 0, stream>>>(feat, fc1_w, fc1out);
  k_fc2_rigid<<<1, 64, 0, stream>>>(fc1out, fc2_w, fc2_b, RT);
  k_grid<<<dim3(512, 8), 256, 0, stream>>>(RT, grd);
  k_sample<<<dim3(512, 8), 256, 0, stream>>>(x, grd, Xt);
  k_moved_reg<<<8, 128, 0, stream>>>(pos, centers, grd, moved, reg);
}